// APPNP_22660247453733
// MI455X (gfx1250) — compile-verified
//
#include <hip/hip_runtime.h>
#include <hip/hip_bf16.h>
#include <cmath>

#define N_NODES   50000
#define N_EDGES   800000
#define IN_DIM    512
#define HIDDEN    64
#define N_CLASSES 40
#define K_LAYERS  10
#define ALPHA     0.1f

// LDS row stride for the 16x512 A tile: 516 % 64 == 4 -> the 16 lanes that
// read A fragments (rows 0..15, same column) land on distinct banks.
#define LDS_STRIDE 516

typedef __attribute__((ext_vector_type(2))) float v2f;
typedef __attribute__((ext_vector_type(8))) float v8f;
typedef __attribute__((ext_vector_type(4))) int   v4i;

typedef __attribute__((address_space(1))) v4i* glob_v4i_ptr;
typedef __attribute__((address_space(3))) v4i* lds_v4i_ptr;

#if defined(__gfx1250__) && __has_builtin(__builtin_amdgcn_global_load_async_to_lds_b128)
#define USE_ASYNC_LDS 1
#else
#define USE_ASYNC_LDS 0
#endif

// ---------------------------------------------------------------------------
// Kernel 1: h0 = relu(feature @ W1 + b1);  h = h0
// One block = one 16-row tile of the output; its 4 waves cover the 4 column
// tiles (HIDDEN/16).  The 16x512 A tile is staged once per block into LDS
// (async-to-LDS when available), then each wave runs 128 chained
// V_WMMA_F32_16X16X4_F32 ops reading A fragments from LDS.
// ---------------------------------------------------------------------------
__global__ __launch_bounds__(128) void gemm1_wmma(
    const float* __restrict__ feature,  // [N_NODES, IN_DIM]
    const float* __restrict__ W1,       // [IN_DIM, HIDDEN]
    const float* __restrict__ b1,       // [HIDDEN]
    float* __restrict__ h0,             // [N_NODES, HIDDEN]
    float* __restrict__ h)              // [N_NODES, HIDDEN]
{
    __shared__ float ldsA[16 * LDS_STRIDE];   // ~33 KB of the 320 KB WGP LDS

    const int mTile = blockIdx.x;             // 3125 row tiles
    const int nTile = threadIdx.x >> 5;       // wave id = column tile (0..3)
    const int lane  = threadIdx.x & 31;
    const int tid   = threadIdx.x;

    // ---- Stage A tile: 16 rows x 512 cols = 2048 float4, 16 per thread ----
    const float* gA = feature + (size_t)mTile * 16 * IN_DIM;
#pragma unroll
    for (int i = 0; i < 16; ++i) {
        const int idx = tid + 128 * i;        // float4 index
        const int r   = idx >> 7;             // row 0..15
        const int c4  = idx & 127;            // float4 column 0..127
        const float* gp = gA + (size_t)r * IN_DIM + 4 * c4;
        float*       lp = &ldsA[r * LDS_STRIDE + 4 * c4];
#if USE_ASYNC_LDS
        __builtin_amdgcn_global_load_async_to_lds_b128(
            (glob_v4i_ptr)gp, (lds_v4i_ptr)lp, 0, 0);
#else
        *(float4*)lp = *(const float4*)gp;
#endif
    }
#if USE_ASYNC_LDS
#if __has_builtin(__builtin_amdgcn_s_wait_asynccnt)
    __builtin_amdgcn_s_wait_asynccnt(0);
#else
    asm volatile("s_wait_asynccnt 0x0" ::: "memory");
#endif
#endif
    __syncthreads();

    // ---- WMMA main loop ----
    const int hi  = lane >> 4;                // half-wave select
    const int l15 = lane & 15;
    const int col = nTile * 16 + l15;         // B/C column for this lane

    // A frag from LDS: lane holds ldsA[l15][k + 2*hi + j], j=0,1 (b64 ds_load)
    const float* aLds  = &ldsA[l15 * LDS_STRIDE + 2 * hi];
    // B frag from global (W1 is 128 KB, L2/L0 resident): W1[k + 2*hi + j][col]
    const float* bBase = W1 + (size_t)(2 * hi) * HIDDEN + col;

    v8f acc = {};
    for (int k = 0; k < IN_DIM; k += 4) {
        v2f a;
        a.x = aLds[k + 0];
        a.y = aLds[k + 1];
        v2f b;
        b.x = bBase[(size_t)(k + 0) * HIDDEN];
        b.y = bBase[(size_t)(k + 1) * HIDDEN];
        // 8 args: (neg_a, A, neg_b, B, c_mod, C, reuse_a, reuse_b)
        acc = __builtin_amdgcn_wmma_f32_16x16x4_f32(
            false, a, false, b, (short)0, acc, false, false);
    }

    // ---- Epilogue: C/D layout -> VGPR r: M = mTile*16 + r + 8*hi, N = col --
    const float bias = b1[col];
#pragma unroll
    for (int r = 0; r < 8; ++r) {
        const int orow = mTile * 16 + r + 8 * hi;
        float v = acc[r] + bias;
        v = v > 0.0f ? v : 0.0f;
        h0[(size_t)orow * HIDDEN + col] = v;
        h [(size_t)orow * HIDDEN + col] = v;
    }
}

// ---------------------------------------------------------------------------
// Zero the aggregation buffer (float4 vectorized) -- run once, before loop.
// ---------------------------------------------------------------------------
__global__ __launch_bounds__(256) void zero_kernel(float4* __restrict__ p, int n4)
{
    const int i = blockIdx.x * blockDim.x + threadIdx.x;
    if (i < n4) p[i] = make_float4(0.0f, 0.0f, 0.0f, 0.0f);
}

// ---------------------------------------------------------------------------
// Edge scatter: agg[dst] += h[src] * w   (16 threads per edge, float4 each).
// h/agg are L2-resident (12.8 MB each vs 192 MB L2); hardware f32 atomics.
// ---------------------------------------------------------------------------
__global__ __launch_bounds__(256) void edge_scatter(
    const int*   __restrict__ src,
    const int*   __restrict__ dst,
    const float* __restrict__ w,
    const float* __restrict__ h,
    float*       __restrict__ agg)
{
    const long long tid = (long long)blockIdx.x * blockDim.x + threadIdx.x;
    if (tid >= (long long)N_EDGES * 16) return;
    const int e = (int)(tid >> 4);
    const int c = ((int)tid & 15) << 2;

    // stream-ahead prefetch of the edge list (global_prefetch_b8)
    if (e + 8192 < N_EDGES) {
        __builtin_prefetch(src + e + 8192, 0, 0);
        __builtin_prefetch(dst + e + 8192, 0, 0);
    }

    const int   s  = src[e];
    const int   d  = dst[e];
    const float wt = w[e];

    const float4 v = *(const float4*)(h + (size_t)s * HIDDEN + c);
    float* a = agg + (size_t)d * HIDDEN + c;
    unsafeAtomicAdd(a + 0, v.x * wt);
    unsafeAtomicAdd(a + 1, v.y * wt);
    unsafeAtomicAdd(a + 2, v.z * wt);
    unsafeAtomicAdd(a + 3, v.w * wt);
}

// ---------------------------------------------------------------------------
// h = (1-alpha)*agg + alpha*h0, and re-zero agg for the next iteration
// (each float4 of agg is consumed by exactly one thread -> race-free).
// ---------------------------------------------------------------------------
__global__ __launch_bounds__(256) void combine_zero_kernel(
    float4*       __restrict__ agg,
    const float4* __restrict__ h0,
    float4*       __restrict__ h,
    int n4)
{
    const int i = blockIdx.x * blockDim.x + threadIdx.x;
    if (i >= n4) return;
    const float4 a = agg[i];
    const float4 z = h0[i];
    float4 r;
    r.x = (1.0f - ALPHA) * a.x + ALPHA * z.x;
    r.y = (1.0f - ALPHA) * a.y + ALPHA * z.y;
    r.z = (1.0f - ALPHA) * a.z + ALPHA * z.z;
    r.w = (1.0f - ALPHA) * a.w + ALPHA * z.w;
    h[i]   = r;
    agg[i] = make_float4(0.0f, 0.0f, 0.0f, 0.0f);
}

// ---------------------------------------------------------------------------
// Head: logits = h @ W2 + b2 ; out = log_softmax(logits).  One thread/node.
// (256 MFLOP total -- negligible next to the propagation loop.)
// ---------------------------------------------------------------------------
__global__ __launch_bounds__(256) void head_kernel(
    const float* __restrict__ h,    // [N_NODES, HIDDEN]
    const float* __restrict__ W2,   // [HIDDEN, N_CLASSES]
    const float* __restrict__ b2,   // [N_CLASSES]
    float*       __restrict__ out)  // [N_NODES, N_CLASSES]
{
    const int node = blockIdx.x * blockDim.x + threadIdx.x;
    if (node >= N_NODES) return;

    float hr[HIDDEN];
    const float4* hp = (const float4*)(h + (size_t)node * HIDDEN);
#pragma unroll
    for (int k = 0; k < HIDDEN / 4; ++k) {
        const float4 t = hp[k];
        hr[4 * k + 0] = t.x;
        hr[4 * k + 1] = t.y;
        hr[4 * k + 2] = t.z;
        hr[4 * k + 3] = t.w;
    }

    float lg[N_CLASSES];
    float m = -INFINITY;
    for (int c = 0; c < N_CLASSES; ++c) {
        float s = b2[c];
#pragma unroll
        for (int k = 0; k < HIDDEN; ++k)
            s = fmaf(hr[k], W2[(size_t)k * N_CLASSES + c], s);
        lg[c] = s;
        m = fmaxf(m, s);
    }
    float sum = 0.0f;
    for (int c = 0; c < N_CLASSES; ++c) sum += __expf(lg[c] - m);
    const float lse = m + __logf(sum);
    for (int c = 0; c < N_CLASSES; ++c)
        out[(size_t)node * N_CLASSES + c] = lg[c] - lse;
}

// ---------------------------------------------------------------------------
extern "C" void kernel_launch(void* const* d_in, const int* in_sizes, int n_in,
                              void* d_out, int out_size, void* d_ws, size_t ws_size,
                              hipStream_t stream)
{
    const float* feature = (const float*)d_in[0];
    const int*   esrc    = (const int*)  d_in[1];
    const int*   edst    = (const int*)  d_in[2];
    const float* ew      = (const float*)d_in[3];
    const float* W1      = (const float*)d_in[4];
    const float* b1      = (const float*)d_in[5];
    const float* W2      = (const float*)d_in[6];
    const float* b2      = (const float*)d_in[7];
    float* out = (float*)d_out;

    const size_t matBytes = (size_t)N_NODES * HIDDEN * sizeof(float); // 12.8 MB
    char* ws   = (char*)d_ws;
    float* h0  = (float*)(ws);
    float* h   = (float*)(ws + matBytes);
    float* agg = (float*)(ws + 2 * matBytes);

    // GEMM1 + ReLU: one block per 16-row tile, 4 waves = 4 column tiles.
    gemm1_wmma<<<N_NODES / 16, 128, 0, stream>>>(feature, W1, b1, h0, h);

    const int n4 = N_NODES * HIDDEN / 4;                 // 800000 float4
    const long long edgeWork = (long long)N_EDGES * 16;  // 12.8M threads
    const int edgeBlocks = (int)((edgeWork + 255) / 256);

    // agg = 0 once; combine_zero re-zeroes it every iteration thereafter.
    zero_kernel<<<(n4 + 255) / 256, 256, 0, stream>>>((float4*)agg, n4);

    for (int it = 0; it < K_LAYERS; ++it) {
        edge_scatter<<<edgeBlocks, 256, 0, stream>>>(esrc, edst, ew, h, agg);
        combine_zero_kernel<<<(n4 + 255) / 256, 256, 0, stream>>>(
            (float4*)agg, (const float4*)h0, (float4*)h, n4);
    }

    head_kernel<<<(N_NODES + 255) / 256, 256, 0, stream>>>(h, W2, b2, out);
}